// GGAHR2HK_24979529793892
// MI455X (gfx1250) — compile-verified
//
#include <hip/hip_runtime.h>
#include <math.h>

#define NSO   18   // spin-expanded orbitals per atom
#define FEAT  58   // orbpair feature dim
#define TILE  16   // edges per block (matches WMMA N)

typedef __attribute__((ext_vector_type(2))) float v2f;
typedef __attribute__((ext_vector_type(8))) float v8f;

// feature index -> (p, q, fac) in the 9x9 block  (s:0, p:1-3, d:4-8)
__constant__ int c_p[FEAT] = {
  0,
  0,0,0,
  0,0,0,0,0,
  1,1,1, 2,2,2, 3,3,3,
  1,1,1,1,1, 2,2,2,2,2, 3,3,3,3,3,
  4,4,4,4,4, 5,5,5,5,5, 6,6,6,6,6, 7,7,7,7,7, 8,8,8,8,8
};
__constant__ int c_q[FEAT] = {
  0,
  1,2,3,
  4,5,6,7,8,
  1,2,3, 1,2,3, 1,2,3,
  4,5,6,7,8, 4,5,6,7,8, 4,5,6,7,8,
  4,5,6,7,8, 4,5,6,7,8, 4,5,6,7,8, 4,5,6,7,8, 4,5,6,7,8
};
__constant__ float c_fac[FEAT] = {
  0.5f,
  1.f,1.f,1.f,
  1.f,1.f,1.f,1.f,1.f,
  0.5f,0.5f,0.5f, 0.5f,0.5f,0.5f, 0.5f,0.5f,0.5f,
  1.f,1.f,1.f,1.f,1.f, 1.f,1.f,1.f,1.f,1.f, 1.f,1.f,1.f,1.f,1.f,
  0.5f,0.5f,0.5f,0.5f,0.5f, 0.5f,0.5f,0.5f,0.5f,0.5f,
  0.5f,0.5f,0.5f,0.5f,0.5f, 0.5f,0.5f,0.5f,0.5f,0.5f,
  0.5f,0.5f,0.5f,0.5f,0.5f
};

static __device__ __forceinline__ void atomic_add_f32(float* p, float v) {
  __hip_atomic_fetch_add(p, v, __ATOMIC_RELAXED, __HIP_MEMORY_SCOPE_AGENT);
}

// ---------------------------------------------------------------------------
// 1) zero-fill the 340MB output with b128 stores (bandwidth-bound phase)
// ---------------------------------------------------------------------------
__global__ void zero_kernel(float* __restrict__ out, long n) {
  long n4 = n >> 2;
  float4* o4 = (float4*)out;
  long i = (long)blockIdx.x * blockDim.x + threadIdx.x;
  long stride = (long)gridDim.x * blockDim.x;
  float4 z = {0.f, 0.f, 0.f, 0.f};
  for (long t = i; t < n4; t += stride) o4[t] = z;
  long tail = (n4 << 2) + i;
  if (tail < n) out[tail] = 0.f;
}

// ---------------------------------------------------------------------------
// 2) onsite blocks: real, phase = 1 for every k; add M at (a,b) and (b,a)
// ---------------------------------------------------------------------------
__global__ void onsite_kernel(const float* __restrict__ feat,
                              float* __restrict__ out, int N, int K) {
  int atom = blockIdx.x;
  int ld = N * NSO;
  long ld2 = (long)ld * ld;
  for (int f = threadIdx.x; f < FEAT; f += blockDim.x) {
    float M = c_fac[f] * feat[atom * FEAT + f];
    long rowb = (long)atom * NSO + 2 * c_p[f];
    long colb = (long)atom * NSO + 2 * c_q[f];
    for (int k = 0; k < K; ++k) {
      long kb = (long)k * ld2;
      #pragma unroll
      for (int s = 0; s < 2; ++s) {
        long a = rowb + s, b = colb + s;
        atomic_add_f32(&out[2 * (kb + a * ld + b)], M);
        atomic_add_f32(&out[2 * (kb + b * ld + a)], M);
      }
    }
  }
}

// ---------------------------------------------------------------------------
// 3) edge kernel: theta tile via V_WMMA_F32_16X16X4_F32, then sincos + scatter
//    A (16x4) = 2*pi*kpoints rows (zero-padded), B (4x16) = edge shifts
// ---------------------------------------------------------------------------
__global__ void edge_kernel(const float* __restrict__ feat,
                            const int*   __restrict__ eidx,
                            const float* __restrict__ shift,
                            const float* __restrict__ kpts,
                            float* __restrict__ out,
                            int N, int E, int K) {
  __shared__ float s_theta[16 * 16];
  __shared__ float s_phre[16 * TILE];
  __shared__ float s_phim[16 * TILE];
  __shared__ float s_feat[TILE * 60];   // padded stride
  __shared__ int   s_i[TILE], s_j[TILE];

  const int e0 = blockIdx.x * TILE;
  const int nvalid = (E - e0 < TILE) ? (E - e0) : TILE;
  const int tid = threadIdx.x;
  const int lane = tid & 31;
  const int lo = lane & 15;
  const int hi = lane >> 4;
  const float TWO_PI = 6.283185307179586f;

  // --- WMMA operands per documented f32 16x16x4 layouts ---
  // A: lane<16 holds (K=0,K=1) of row M=lo; lane>=16 holds (K=2,K=3)
  v2f a; a.x = 0.f; a.y = 0.f;
  if (lo < K) {
    if (!hi) { a.x = TWO_PI * kpts[lo * 3 + 0]; a.y = TWO_PI * kpts[lo * 3 + 1]; }
    else     { a.x = TWO_PI * kpts[lo * 3 + 2]; a.y = 0.f; }
  }
  // B: lane<16 holds (K=0,K=1) of col N=lo; lane>=16 holds (K=2,K=3)
  v2f b; b.x = 0.f; b.y = 0.f;
  if (lo < nvalid) {
    if (!hi) { b.x = shift[(e0 + lo) * 3 + 0]; b.y = shift[(e0 + lo) * 3 + 1]; }
    else     { b.x = shift[(e0 + lo) * 3 + 2]; b.y = 0.f; }
  }
  v8f c = {};
  // All waves execute (EXEC all-ones, no divergence around WMMA); wave 0 stores.
  v8f d = __builtin_amdgcn_wmma_f32_16x16x4_f32(
      false, a, false, b, (short)0, c, false, false);
  if (tid < 32) {
    #pragma unroll
    for (int v = 0; v < 8; ++v) {
      int m = v + (hi ? 8 : 0);            // C/D layout: VGPR v -> row v / v+8
      s_theta[m * 16 + lo] = d[v];
    }
  }
  __syncthreads();

  // phases
  if (tid < K * TILE) {
    int k = tid / TILE, e = tid % TILE;
    float sn, cs;
    __sincosf(s_theta[k * 16 + e], &sn, &cs);
    s_phre[k * TILE + e] = cs;
    s_phim[k * TILE + e] = sn;
  }
  // stage features + edge endpoints
  for (int idx = tid; idx < nvalid * FEAT; idx += blockDim.x) {
    int e = idx / FEAT, f = idx % FEAT;
    s_feat[e * 60 + f] = feat[(long)(e0 + e) * FEAT + f];
  }
  if (tid < nvalid) { s_i[tid] = eidx[e0 + tid]; s_j[tid] = eidx[E + e0 + tid]; }
  __syncthreads();

  const int ld = N * NSO;
  const long ld2 = (long)ld * ld;
  for (int idx = tid; idx < nvalid * FEAT; idx += blockDim.x) {
    int e = idx / FEAT, f = idx % FEAT;
    float M = c_fac[f] * s_feat[e * 60 + f];
    long rowb = (long)s_i[e] * NSO + 2 * c_p[f];
    long colb = (long)s_j[e] * NSO + 2 * c_q[f];
    for (int k = 0; k < K; ++k) {
      float pr = s_phre[k * TILE + e], pi = s_phim[k * TILE + e];
      float re = pr * M, im = pi * M;
      long kb = (long)k * ld2;
      #pragma unroll
      for (int s = 0; s < 2; ++s) {
        long aa = rowb + s, bb = colb + s;
        long i1 = 2 * (kb + aa * ld + bb);   // H0[a,b]        += phase*M
        long i2 = 2 * (kb + bb * ld + aa);   // conj(H0^T)[b,a] += conj(phase)*M
        atomic_add_f32(&out[i1],     re);
        atomic_add_f32(&out[i1 + 1], im);
        atomic_add_f32(&out[i2],     re);
        atomic_add_f32(&out[i2 + 1], -im);
      }
    }
  }
}

// ---------------------------------------------------------------------------
extern "C" void kernel_launch(void* const* d_in, const int* in_sizes, int n_in,
                              void* d_out, int out_size, void* d_ws, size_t ws_size,
                              hipStream_t stream) {
  const float* hop   = (const float*)d_in[0];   // [E, 58]
  const float* ons   = (const float*)d_in[1];   // [N, 58]
  const int*   eidx  = (const int*)  d_in[2];   // [2, E]
  const float* shift = (const float*)d_in[3];   // [E, 3]
  const float* kpts  = (const float*)d_in[4];   // [K, 3]
  float* out = (float*)d_out;                   // [K, N*18, N*18] complex64 -> 2 floats

  int E = in_sizes[0] / FEAT;
  int N = in_sizes[1] / FEAT;
  int K = in_sizes[4] / 3;

  long total = (long)out_size;                  // floats
  zero_kernel<<<2048, 256, 0, stream>>>(out, total);
  onsite_kernel<<<N, 64, 0, stream>>>(ons, out, N, K);
  int nblk = (E + TILE - 1) / TILE;
  edge_kernel<<<nblk, 256, 0, stream>>>(hop, eidx, shift, kpts, out, N, E, K);
}